// CustomTGNMemory_87763361726821
// MI455X (gfx1250) — compile-verified
//
#include <hip/hip_runtime.h>
#include <hip/hip_bf16.h>
#include <math.h>
#include <stdint.h>

#define MDIM 128
#define RDIM 172
#define TDIM 128
#define KIN  (2 * MDIM + RDIM + TDIM)   // 556
// A-tile LDS stride (floats): 564*4 = 2256 B -> rows 16B aligned for B128
// async copies, and 564 mod 64 = 52 (gcd 4) -> 16 consecutive rows hit 16
// distinct banks for the WMMA A-fragment ds reads.
#define APAD 564
#define HPAD 132
#define MTILE_MSG 64     // events per block (4 row tiles of 16)
#define MTILE_GRU 32     // nodes per block (2 row tiles of 16)

typedef float v2f __attribute__((ext_vector_type(2)));
typedef float v8f __attribute__((ext_vector_type(8)));

// D = A(16x4 f32) * B(4x16 f32) + C(16x16 f32), full fp32 precision
__device__ __forceinline__ v8f wmma4(v2f a, v2f b, v8f c) {
  return __builtin_amdgcn_wmma_f32_16x16x4_f32(false, a, false, b, (short)0, c,
                                               false, false);
}

__device__ __forceinline__ float sigmoidf_(float x) {
  return 1.0f / (1.0f + expf(-x));
}

// CDNA5 async copy: copies 16 bytes from the global address gptr to the LDS address lds_off; tracked by ASYNCcnt
__device__ __forceinline__ void async_copy_b128(uint32_t lds_off,
                                                const void* gptr) {
  asm volatile("global_load_async_to_lds_b128 %0, %1, off"
               :: "v"(lds_off), "v"((uint64_t)(uintptr_t)gptr)
               : "memory");
}
__device__ __forceinline__ void wait_asynccnt0() {
  asm volatile("s_wait_asynccnt 0" ::: "memory");
}

// ---------------------------------------------------------------------------
// Kernel 0: zero accumulators (sum_msg + t_max live in d_out, cnt in d_ws)
// ---------------------------------------------------------------------------
__global__ void tgn_zero_kernel(float* __restrict__ sum_msg,
                                float* __restrict__ tmax,
                                float* __restrict__ cnt, int N) {
  int i = blockIdx.x * blockDim.x + threadIdx.x;
  if (i < N * MDIM) sum_msg[i] = 0.0f;
  if (i < N) { tmax[i] = 0.0f; cnt[i] = 0.0f; }
}

// ---------------------------------------------------------------------------
// Kernel 0b: transpose msg_w [556,128] -> mwT [128,556] so both K-values of a
// B fragment are adjacent (single global_load_b64 per fragment).
// ---------------------------------------------------------------------------
__global__ void tgn_transpose_kernel(const float* __restrict__ mw,
                                     float* __restrict__ mwT) {
  int i = blockIdx.x * blockDim.x + threadIdx.x;
  if (i < KIN * MDIM) {
    int k = i / MDIM, c = i % MDIM;
    mwT[(size_t)c * KIN + k] = mw[i];
  }
}

// ---------------------------------------------------------------------------
// Kernel 1: msg = relu(concat(mem[src], mem[dst], raw, cos(dt*w+b)) @ W + b)
// scattered into sum_msg; cnt / t_max accumulated per dst.
// Block = 64 events (4 row tiles). Wave w owns output cols [16w,16w+16) and
// keeps 4 accumulators so each B fragment feeds 4 WMMAs.
// ---------------------------------------------------------------------------
__global__ void __launch_bounds__(256, 4)
tgn_msg_kernel(const int* __restrict__ src, const int* __restrict__ dst,
               const float* __restrict__ t, const float* __restrict__ raw_msg,
               const float* __restrict__ memory,
               const float* __restrict__ last_update,
               const float* __restrict__ te_w, const float* __restrict__ te_b,
               const float* __restrict__ mwT, const float* __restrict__ msg_b,
               float* __restrict__ sum_msg, float* __restrict__ cnt,
               unsigned int* __restrict__ tmax_bits) {
  __shared__ float As[MTILE_MSG * APAD];
  __shared__ int dstL[MTILE_MSG];

  const int tid   = threadIdx.x;
  const int ebase = blockIdx.x * MTILE_MSG;

  // ---- stage A tile (64 x 556). 4 threads per event row. ----
  {
    const int r = tid >> 2;   // event row 0..63
    const int l = tid & 3;    // 4 threads per row
    const int e = ebase + r;
    const int s_n = src[e];
    const int d_n = dst[e];
    const float tv = t[e];
    const float dt = tv - last_update[d_n];
    if (l == 0) {
      dstL[r] = d_n;
      atomicAdd(&cnt[d_n], 1.0f);
      atomicMax(&tmax_bits[d_n], __float_as_uint(tv));  // t >= 0: bit order ok
    }
    const float* mrow_s = memory + (size_t)s_n * MDIM;
    const float* mrow_d = memory + (size_t)d_n * MDIM;
    const float* rrow   = raw_msg + (size_t)e * RDIM;
    const uint32_t lbase = (uint32_t)(uintptr_t)&As[r * APAD];

    // mem[src]: 32 x B128 chunks, 8 per thread (async DMA to LDS)
#pragma unroll
    for (int j = 0; j < 8; ++j) {
      const int c = 4 * (l + 4 * j);
      async_copy_b128(lbase + 4u * c, mrow_s + c);
    }
    // mem[dst]: 32 chunks -> LDS offset +128 floats
#pragma unroll
    for (int j = 0; j < 8; ++j) {
      const int c = 4 * (l + 4 * j);
      async_copy_b128(lbase + 4u * (MDIM + c), mrow_d + c);
    }
    // raw_msg: 43 chunks (172 floats, rows 16B aligned: 172*4=688=16*43)
#pragma unroll
    for (int j = 0; j < 11; ++j) {
      const int ch = l + 4 * j;
      if (ch < RDIM / 4) {
        const int c = 4 * ch;
        async_copy_b128(lbase + 4u * (2 * MDIM + c), rrow + c);
      }
    }
    // dt_enc: computed, 32 cols per thread, plain LDS stores
    for (int j = l; j < TDIM; j += 4) {
      As[r * APAD + 2 * MDIM + RDIM + j] = cosf(dt * te_w[j] + te_b[j]);
    }
    wait_asynccnt0();
  }
  __syncthreads();

  // ---- WMMA: 4 row tiles x 1 col tile per wave, K = 556 step 4 ----
  const int lane = tid & 31;
  const int koff = (lane >> 4) ? 2 : 0;     // upper lane-half holds K+2,K+3
  const int lrow = lane & 15;
  const int col  = (tid >> 5) * 16 + lrow;

  // Per-lane base pointers: all k-offsets become pure immediates.
  const float* A0 = &As[(0 * 16 + lrow) * APAD + koff];
  const float* A1 = &As[(1 * 16 + lrow) * APAD + koff];
  const float* A2 = &As[(2 * 16 + lrow) * APAD + koff];
  const float* A3 = &As[(3 * 16 + lrow) * APAD + koff];
  const float* Bq = mwT + (size_t)col * KIN + koff;

  v8f acc0 = (v8f){0.f,0.f,0.f,0.f,0.f,0.f,0.f,0.f};
  v8f acc1 = acc0, acc2 = acc0, acc3 = acc0;

#pragma unroll 2
  for (int k = 0; k < KIN; k += 4) {
    v2f b  = { Bq[k], Bq[k + 1] };          // single b64, immediate offset
    v2f a0 = { A0[k], A0[k + 1] };
    v2f a1 = { A1[k], A1[k + 1] };
    v2f a2 = { A2[k], A2[k + 1] };
    v2f a3 = { A3[k], A3[k + 1] };
    acc0 = wmma4(a0, b, acc0);
    acc1 = wmma4(a1, b, acc1);
    acc2 = wmma4(a2, b, acc2);
    acc3 = wmma4(a3, b, acc3);
  }

  // bias + relu + scatter-add into sum_msg[dst]
  const float bias = msg_b[col];
  v8f accs[4] = { acc0, acc1, acc2, acc3 };
#pragma unroll
  for (int tt = 0; tt < 4; ++tt) {
#pragma unroll
    for (int r = 0; r < 8; ++r) {
      const int m = tt * 16 + r + (koff ? 8 : 0);   // event row in block
      float v = fmaxf(accs[tt][r] + bias, 0.0f);
      atomicAdd(&sum_msg[(size_t)dstL[m] * MDIM + col], v);
    }
  }
}

// ---------------------------------------------------------------------------
// Kernel 2: GRU update, 32 nodes per block (2 row tiles). Wave w owns hidden
// cols [16w,16w+16); 12 accumulators (r/z/n gates x {input,hidden} x 2 tiles)
// so each B fragment feeds 2 WMMAs. Reads sum_msg from d_out and overwrites
// it in place with new_memory.
// ---------------------------------------------------------------------------
__global__ void __launch_bounds__(256, 4)
tgn_gru_kernel(const float* __restrict__ memory,
               const float* __restrict__ last_update,
               const float* __restrict__ w_ih, const float* __restrict__ w_hh,
               const float* __restrict__ b_ih, const float* __restrict__ b_hh,
               const float* __restrict__ cnt,
               float* __restrict__ out_mem,   // in: sum_msg, out: new_memory
               float* __restrict__ out_lu) {  // in: t_max,  out: new_last_update
  __shared__ float Ag[MTILE_GRU * HPAD];
  __shared__ float Hm[MTILE_GRU * HPAD];
  __shared__ float cntL[MTILE_GRU];

  const int tid = threadIdx.x;
  const int nb  = blockIdx.x * MTILE_GRU;

  // ---- stage agg (sum/max(cnt,1)) and h tiles; float4 traffic ----
  {
    const int r = tid >> 3;    // node row 0..31
    const int l = tid & 7;     // 8 threads per row
    const int node = nb + r;
    const float cv = cnt[node];
    if (l == 0) cntL[r] = cv;
    const float inv = 1.0f / fmaxf(cv, 1.0f);
#pragma unroll
    for (int j = 0; j < 4; ++j) {
      const int c = 4 * (l + 8 * j);
      const float4 s4 = *(const float4*)&out_mem[(size_t)node * MDIM + c];
      const float4 h4 = *(const float4*)&memory[(size_t)node * MDIM + c];
      *(float4*)&Ag[r * HPAD + c] =
          make_float4(s4.x * inv, s4.y * inv, s4.z * inv, s4.w * inv);
      *(float4*)&Hm[r * HPAD + c] = h4;
    }
  }
  __syncthreads();

  const int lane = tid & 31;
  const int koff = (lane >> 4) ? 2 : 0;
  const int lrow = lane & 15;
  const int col  = (tid >> 5) * 16 + lrow;  // 0..127

  // Per-lane base pointers (immediate k-offsets in the loop)
  const float* Ar0 = &Ag[lrow * HPAD + koff];
  const float* Ar1 = &Ag[(16 + lrow) * HPAD + koff];
  const float* Hr0 = &Hm[lrow * HPAD + koff];
  const float* Hr1 = &Hm[(16 + lrow) * HPAD + koff];
  // w_ih/w_hh are [3H, H] row-major; B(k, j=g*128+col) = w[(g*128+col)*128+k]
  const float* Wi0 = w_ih + (size_t)col * MDIM + koff;
  const float* Wi1 = w_ih + (size_t)(MDIM + col) * MDIM + koff;
  const float* Wi2 = w_ih + (size_t)(2 * MDIM + col) * MDIM + koff;
  const float* Wh0 = w_hh + (size_t)col * MDIM + koff;
  const float* Wh1 = w_hh + (size_t)(MDIM + col) * MDIM + koff;
  const float* Wh2 = w_hh + (size_t)(2 * MDIM + col) * MDIM + koff;

  v8f z8 = (v8f){0.f,0.f,0.f,0.f,0.f,0.f,0.f,0.f};
  v8f ai0a = z8, ai0b = z8, ai1a = z8, ai1b = z8, ai2a = z8, ai2b = z8;
  v8f ah0a = z8, ah0b = z8, ah1a = z8, ah1b = z8, ah2a = z8, ah2b = z8;

#pragma unroll 2
  for (int k = 0; k < MDIM; k += 4) {
    v2f a0 = { Ar0[k], Ar0[k + 1] };
    v2f a1 = { Ar1[k], Ar1[k + 1] };
    v2f h0 = { Hr0[k], Hr0[k + 1] };
    v2f h1 = { Hr1[k], Hr1[k + 1] };
    v2f b;
    b = (v2f){ Wi0[k], Wi0[k + 1] }; ai0a = wmma4(a0, b, ai0a); ai0b = wmma4(a1, b, ai0b);
    b = (v2f){ Wi1[k], Wi1[k + 1] }; ai1a = wmma4(a0, b, ai1a); ai1b = wmma4(a1, b, ai1b);
    b = (v2f){ Wi2[k], Wi2[k + 1] }; ai2a = wmma4(a0, b, ai2a); ai2b = wmma4(a1, b, ai2b);
    b = (v2f){ Wh0[k], Wh0[k + 1] }; ah0a = wmma4(h0, b, ah0a); ah0b = wmma4(h1, b, ah0b);
    b = (v2f){ Wh1[k], Wh1[k + 1] }; ah1a = wmma4(h0, b, ah1a); ah1b = wmma4(h1, b, ah1b);
    b = (v2f){ Wh2[k], Wh2[k + 1] }; ah2a = wmma4(h0, b, ah2a); ah2b = wmma4(h1, b, ah2b);
  }

  const float bir  = b_ih[col],            bhr = b_hh[col];
  const float biz  = b_ih[MDIM + col],     bhz = b_hh[MDIM + col];
  const float bin_ = b_ih[2 * MDIM + col], bhn = b_hh[2 * MDIM + col];

  v8f ai[3][2] = { {ai0a, ai0b}, {ai1a, ai1b}, {ai2a, ai2b} };
  v8f ah[3][2] = { {ah0a, ah0b}, {ah1a, ah1b}, {ah2a, ah2b} };

#pragma unroll
  for (int tt = 0; tt < 2; ++tt) {
#pragma unroll
    for (int r = 0; r < 8; ++r) {
      const int m = tt * 16 + r + (koff ? 8 : 0);
      const float rg = sigmoidf_((ai[0][tt][r] + bir) + (ah[0][tt][r] + bhr));
      const float zg = sigmoidf_((ai[1][tt][r] + biz) + (ah[1][tt][r] + bhz));
      const float ng = tanhf((ai[2][tt][r] + bin_) + rg * (ah[2][tt][r] + bhn));
      const float h  = Hm[m * HPAD + col];
      const float upd = (1.0f - zg) * ng + zg * h;
      out_mem[(size_t)(nb + m) * MDIM + col] = (cntL[m] > 0.0f) ? upd : h;
    }
  }

  if (tid < MTILE_GRU) {
    const int node = nb + tid;
    const float tm = out_lu[node];  // accumulated t_max (bits == float order)
    out_lu[node] = (cntL[tid] > 0.0f) ? tm : last_update[node];
  }
}

// ---------------------------------------------------------------------------
extern "C" void kernel_launch(void* const* d_in, const int* in_sizes, int n_in,
                              void* d_out, int out_size, void* d_ws,
                              size_t ws_size, hipStream_t stream) {
  const int*   src = (const int*)  d_in[0];
  const int*   dst = (const int*)  d_in[1];
  const float* t   = (const float*)d_in[2];
  const float* raw = (const float*)d_in[3];
  const float* mem = (const float*)d_in[4];
  const float* lu  = (const float*)d_in[5];
  const float* tew = (const float*)d_in[6];
  const float* teb = (const float*)d_in[7];
  const float* mw  = (const float*)d_in[8];
  const float* mb  = (const float*)d_in[9];
  const float* wih = (const float*)d_in[10];
  const float* whh = (const float*)d_in[11];
  const float* bih = (const float*)d_in[12];
  const float* bhh = (const float*)d_in[13];

  const int E = in_sizes[0];
  const int N = in_sizes[5];

  float* out     = (float*)d_out;
  float* sum_msg = out;                        // [N,128]: sum_msg -> new_memory
  float* out_lu  = out + (size_t)N * MDIM;     // [N]: t_max -> new_last_update
  float* cnt     = (float*)d_ws;               // [N]
  float* mwT     = cnt + N;                    // [128,556] transposed weights

  {
    const int blocks = (N * MDIM + 255) / 256;
    tgn_zero_kernel<<<blocks, 256, 0, stream>>>(sum_msg, out_lu, cnt, N);
  }
  {
    const int blocks = (KIN * MDIM + 255) / 256;
    tgn_transpose_kernel<<<blocks, 256, 0, stream>>>(mw, mwT);
  }
  tgn_msg_kernel<<<E / MTILE_MSG, 256, 0, stream>>>(
      src, dst, t, raw, mem, lu, tew, teb, mwT, mb, sum_msg, cnt,
      (unsigned int*)out_lu);
  tgn_gru_kernel<<<N / MTILE_GRU, 256, 0, stream>>>(
      mem, lu, wih, whh, bih, bhh, cnt, sum_msg, out_lu);
}